// Block_48249662603768
// MI455X (gfx1250) — compile-verified
//
#include <hip/hip_runtime.h>
#include <math.h>

// Problem constants (from reference)
#define NMODEL 8
#define NBATCH 4096
#define SEQT   33
#define DMODEL 8
#define DFF    28
#define TP     48        // T padded to 3 WMMA row tiles
#define NW     4         // waves per block (wave32)
#define BLK_X  128       // blocks per model
#define EPSLN  1e-5f

typedef float v2f __attribute__((ext_vector_type(2)));
typedef float v8f __attribute__((ext_vector_type(8)));
typedef int v4i_gcc __attribute__((vector_size(4 * sizeof(int))));  // matches builtin param

#if __has_builtin(__builtin_amdgcn_global_load_async_to_lds_b128)
#define USE_ASYNC_LDS 1
#else
#define USE_ASYNC_LDS 0
#endif

// integer-casts to form AS-qualified pointers (flat LDS addr low 32 bits == LDS offset, ISA 10.2)
#define AS1V4(p) ((__attribute__((address_space(1))) v4i_gcc*)(unsigned long long)(p))
#define AS3V4(p) ((__attribute__((address_space(3))) v4i_gcc*)(unsigned int)(unsigned long long)(p))

// D = A(16x4) * B(4x16) + C(16x16), fp32, wave32
static __device__ __forceinline__ v8f wmma4(v2f a, v2f b, v8f c) {
  return __builtin_amdgcn_wmma_f32_16x16x4_f32(
      /*neg_a=*/false, a, /*neg_b=*/false, b,
      /*c_mod=*/(short)0, c, /*reuse_a=*/false, /*reuse_b=*/false);
}

__global__ void __launch_bounds__(NW * 32)
fused_block_kernel(const float* __restrict__ x,
                   const float* __restrict__ ln1_w, const float* __restrict__ ln1_b,
                   const float* __restrict__ qkv_w, const float* __restrict__ proj_w,
                   const float* __restrict__ ln2_w, const float* __restrict__ ln2_b,
                   const float* __restrict__ fc1_A, const float* __restrict__ fc1_B,
                   const float* __restrict__ fc1_Wf,
                   const float* __restrict__ fc2_A, const float* __restrict__ fc2_B,
                   const float* __restrict__ fc2_Wf,
                   float* __restrict__ out) {
  // ---- LDS: per-model weights (padded for WMMA B-fragments) + per-wave tiles ----
  __shared__ float s_wqkv[8 * 32];    // [d][o] o<24 valid, rest 0
  __shared__ float s_wproj[8 * 16];   // [d][o] o<8 valid
  __shared__ float s_wfc1[8 * 32];    // fc1_A@fc1_B + fc1_Wf, [d][f] f<28 valid
  __shared__ float s_wfc2[28 * 16];   // fc2_A@fc2_B + fc2_Wf, [f][o] o<8 valid
  __shared__ float s_ln1wv[8], s_ln1bv[8], s_ln2wv[8], s_ln2bv[8];
  __shared__ __align__(16) float s_xb [NW][TP * 8];   // x tile (rows >= T stay zero)
  __shared__ __align__(16) float s_hb [NW][TP * 16];  // LN out; reused as P scratch [16][48]
  __shared__ __align__(16) float s_buf[NW][TP * 32];  // qkv (q|k|v|pad); reused as h1
  __shared__ __align__(16) float s_yb [NW][TP * 16];  // attention output

  const int tid  = threadIdx.x;
  const int m    = blockIdx.y;
  const int wave = tid >> 5;
  const int lane = tid & 31;
  const int l16  = lane & 15;
  const int half = lane >> 4;
  const int nthr = NW * 32;

  // ---- build per-model weights in LDS (low-rank terms folded in) ----
  if (tid < 8) {
    s_ln1wv[tid] = ln1_w[m * 8 + tid];  s_ln1bv[tid] = ln1_b[m * 8 + tid];
    s_ln2wv[tid] = ln2_w[m * 8 + tid];  s_ln2bv[tid] = ln2_b[m * 8 + tid];
  }
  for (int i = tid; i < 8 * 32; i += nthr) {
    int d = i >> 5, o = i & 31;
    s_wqkv[i] = (o < 24) ? qkv_w[(m * 8 + d) * 24 + o] : 0.f;
  }
  for (int i = tid; i < 8 * 16; i += nthr) {
    int d = i >> 4, o = i & 15;
    s_wproj[i] = (o < 8) ? proj_w[(m * 8 + d) * 8 + o] : 0.f;
  }
  for (int i = tid; i < 8 * 32; i += nthr) {
    int d = i >> 5, f = i & 31;
    float w = 0.f;
    if (f < DFF) {
      w = fc1_Wf[(m * 8 + d) * DFF + f];
      #pragma unroll
      for (int r = 0; r < 3; ++r)
        w += fc1_A[(m * 8 + d) * 3 + r] * fc1_B[(m * 3 + r) * DFF + f];
    }
    s_wfc1[i] = w;
  }
  for (int i = tid; i < 28 * 16; i += nthr) {
    int f = i >> 4, o = i & 15;
    float w = 0.f;
    if (o < 8) {
      w = fc2_Wf[(m * DFF + f) * 8 + o];
      #pragma unroll
      for (int r = 0; r < 3; ++r)
        w += fc2_A[(m * DFF + f) * 3 + r] * fc2_B[(m * 3 + r) * 8 + o];
    }
    s_wfc2[i] = w;
  }
  __syncthreads();

  float* xb  = s_xb[wave];
  float* hb  = s_hb[wave];
  float* buf = s_buf[wave];
  float* yb  = s_yb[wave];

  // ---- hoist loop-invariant weight B-fragments into registers (17 x v2f) ----
  v2f f_qkv[2][2], f_fc1[2][2], f_proj[2], f_fc2[7];
  #pragma unroll
  for (int nt = 0; nt < 2; ++nt)
    #pragma unroll
    for (int k2 = 0; k2 < 2; ++k2) {
      int kr = k2 * 4 + 2 * half;
      f_qkv[nt][k2] = v2f{ s_wqkv[kr * 32 + nt * 16 + l16],
                           s_wqkv[(kr + 1) * 32 + nt * 16 + l16] };
      f_fc1[nt][k2] = v2f{ s_wfc1[kr * 32 + nt * 16 + l16],
                           s_wfc1[(kr + 1) * 32 + nt * 16 + l16] };
    }
  #pragma unroll
  for (int k2 = 0; k2 < 2; ++k2) {
    int kr = k2 * 4 + 2 * half;
    f_proj[k2] = v2f{ s_wproj[kr * 16 + l16], s_wproj[(kr + 1) * 16 + l16] };
  }
  #pragma unroll
  for (int k2 = 0; k2 < 7; ++k2) {
    int kr = k2 * 4 + 2 * half;
    f_fc2[k2] = v2f{ s_wfc2[kr * 16 + l16], s_wfc2[(kr + 1) * 16 + l16] };
  }

  auto layernorm = [&](const float* src, float* dst, const float* w, const float* bb) {
    for (int t = lane; t < TP; t += 32) {
      float v[8]; float s = 0.f;
      #pragma unroll
      for (int d = 0; d < 8; ++d) { v[d] = src[t * 8 + d]; s += v[d]; }
      float mean = s * 0.125f, ss = 0.f;
      #pragma unroll
      for (int d = 0; d < 8; ++d) { float c = v[d] - mean; ss += c * c; }
      float r = rsqrtf(ss * 0.125f + EPSLN);   // biased variance, matches reference
      #pragma unroll
      for (int d = 0; d < 8; ++d) dst[t * 16 + d] = (v[d] - mean) * r * w[d] + bb[d];
      #pragma unroll
      for (int d = 8; d < 16; ++d) dst[t * 16 + d] = 0.f;
    }
  };

  // zero x-tile pad rows once (never overwritten: residual/store guard t < T)
  for (int i = SEQT * DMODEL + lane; i < TP * DMODEL; i += 32) xb[i] = 0.f;

  const int wgid    = blockIdx.x * NW + wave;  // 0 .. BLK_X*NW-1
  const int wstride = BLK_X * NW;

  for (int b = wgid; b < NBATCH; b += wstride) {
    const float* xg = x   + ((size_t)m * NBATCH + b) * (size_t)(SEQT * DMODEL);
    float*       og = out + ((size_t)m * NBATCH + b) * (size_t)(SEQT * DMODEL);
    if (b + wstride < NBATCH)
      __builtin_prefetch(x + ((size_t)m * NBATCH + b + wstride) * (size_t)(SEQT * DMODEL), 0, 1);

    // ---- load x tile rows 0..32 (66 x float4) ----
#if USE_ASYNC_LDS
    {
      const float4* xg4 = (const float4*)xg;
      float4* xb4 = (float4*)xb;
      #pragma unroll
      for (int it = 0; it < 3; ++it) {
        int i = lane + it * 32;
        if (i < (SEQT * DMODEL) / 4)
          __builtin_amdgcn_global_load_async_to_lds_b128(AS1V4(xg4 + i), AS3V4(xb4 + i), 0, 0);
      }
#if __has_builtin(__builtin_amdgcn_s_wait_asynccnt)
      __builtin_amdgcn_s_wait_asynccnt(0);
      asm volatile("" ::: "memory");
#else
      asm volatile("s_wait_asynccnt 0" ::: "memory");
#endif
    }
#else
    {
      const float4* xg4 = (const float4*)xg;
      float4* xb4 = (float4*)xb;
      #pragma unroll
      for (int it = 0; it < 3; ++it) {
        int i = lane + it * 32;
        if (i < (SEQT * DMODEL) / 4) xb4[i] = xg4[i];
      }
    }
#endif

    // ---- LN1 ----
    layernorm(xb, hb, s_ln1wv, s_ln1bv);

    // ---- QKV: [48x8] @ [8x24->32] ----
    #pragma unroll
    for (int qt = 0; qt < 3; ++qt) {
      #pragma unroll
      for (int nt = 0; nt < 2; ++nt) {
        v8f c = {};
        #pragma unroll
        for (int kc = 0; kc < 8; kc += 4) {
          v2f a = { hb[(qt * 16 + l16) * 16 + kc + 2 * half],
                    hb[(qt * 16 + l16) * 16 + kc + 2 * half + 1] };
          c = wmma4(a, f_qkv[nt][kc >> 2], c);
        }
        #pragma unroll
        for (int i = 0; i < 8; ++i)
          buf[(qt * 16 + i + 8 * half) * 32 + nt * 16 + l16] = c[i];
      }
    }

    // ---- attention: S = QK^T/sqrt(8), causal softmax, Y = P@V ----
    const float sc = 0.35355339059327373f;  // 1/sqrt(8)
    #pragma unroll
    for (int qt = 0; qt < 3; ++qt) {
      v8f sfr[3];
      #pragma unroll
      for (int kt = 0; kt < 3; ++kt) {
        v8f c = {};
        #pragma unroll
        for (int kc = 0; kc < 8; kc += 4) {
          v2f a = { buf[(qt * 16 + l16) * 32 + kc + 2 * half],
                    buf[(qt * 16 + l16) * 32 + kc + 2 * half + 1] };
          v2f bf = { buf[(kt * 16 + l16) * 32 + 8 + kc + 2 * half],     // K^T
                     buf[(kt * 16 + l16) * 32 + 8 + kc + 2 * half + 1] };
          c = wmma4(a, bf, c);
        }
        sfr[kt] = c;
      }
      // scale + causal mask + softmax (vgpr i -> row i+8*half, cols on l16)
      #pragma unroll
      for (int i = 0; i < 8; ++i) {
        int q = qt * 16 + i + 8 * half;
        float mx = -INFINITY;
        #pragma unroll
        for (int kt = 0; kt < 3; ++kt) {
          int kk = kt * 16 + l16;
          float v = sfr[kt][i] * sc;
          v = (kk <= q && kk < SEQT) ? v : -INFINITY;
          sfr[kt][i] = v;
          mx = fmaxf(mx, v);
        }
        #pragma unroll
        for (int off = 8; off >= 1; off >>= 1) mx = fmaxf(mx, __shfl_xor(mx, off, 16));
        float sum = 0.f;
        #pragma unroll
        for (int kt = 0; kt < 3; ++kt) {
          float e = __expf(sfr[kt][i] - mx);
          sfr[kt][i] = e;
          sum += e;
        }
        #pragma unroll
        for (int off = 8; off >= 1; off >>= 1) sum += __shfl_xor(sum, off, 16);
        float inv = 1.f / sum;
        #pragma unroll
        for (int kt = 0; kt < 3; ++kt) sfr[kt][i] *= inv;
      }
      // stage P into LDS (hb reused as [16][48]) for A-fragment reads
      #pragma unroll
      for (int i = 0; i < 8; ++i)
        #pragma unroll
        for (int kt = 0; kt < 3; ++kt)
          hb[(i + 8 * half) * TP + kt * 16 + l16] = sfr[kt][i];
      // Y tile = P[16x48] @ V[48x16(8 valid)]
      v8f cy = {};
      #pragma unroll
      for (int kc = 0; kc < TP; kc += 4) {
        v2f a = { hb[l16 * TP + kc + 2 * half],
                  hb[l16 * TP + kc + 2 * half + 1] };
        v2f bf = { buf[(kc + 2 * half)     * 32 + 16 + l16],
                   buf[(kc + 2 * half + 1) * 32 + 16 + l16] };
        cy = wmma4(a, bf, cy);
      }
      #pragma unroll
      for (int i = 0; i < 8; ++i)
        yb[(qt * 16 + i + 8 * half) * 16 + l16] = cy[i];
    }

    // ---- proj + residual into xb ----
    #pragma unroll
    for (int qt = 0; qt < 3; ++qt) {
      v8f c = {};
      #pragma unroll
      for (int kc = 0; kc < 8; kc += 4) {
        v2f a = { yb[(qt * 16 + l16) * 16 + kc + 2 * half],
                  yb[(qt * 16 + l16) * 16 + kc + 2 * half + 1] };
        c = wmma4(a, f_proj[kc >> 2], c);
      }
      if (l16 < 8) {
        #pragma unroll
        for (int i = 0; i < 8; ++i) {
          int t = qt * 16 + i + 8 * half;
          if (t < SEQT) xb[t * 8 + l16] += c[i];
        }
      }
    }

    // ---- LN2 ----
    layernorm(xb, hb, s_ln2wv, s_ln2bv);

    // ---- fc1 (+folded low-rank) + exact-erf GELU, into buf (stride 32) ----
    #pragma unroll
    for (int qt = 0; qt < 3; ++qt) {
      #pragma unroll
      for (int nt = 0; nt < 2; ++nt) {
        v8f c = {};
        #pragma unroll
        for (int kc = 0; kc < 8; kc += 4) {
          v2f a = { hb[(qt * 16 + l16) * 16 + kc + 2 * half],
                    hb[(qt * 16 + l16) * 16 + kc + 2 * half + 1] };
          c = wmma4(a, f_fc1[nt][kc >> 2], c);
        }
        #pragma unroll
        for (int i = 0; i < 8; ++i) {
          float v = c[i];
          v = 0.5f * v * (1.f + erff(v * 0.7071067811865475f));
          buf[(qt * 16 + i + 8 * half) * 32 + nt * 16 + l16] = v;
        }
      }
    }

    // ---- fc2 (+folded low-rank) + residual, store to global ----
    #pragma unroll
    for (int qt = 0; qt < 3; ++qt) {
      v8f c = {};
      #pragma unroll
      for (int kc = 0; kc < DFF; kc += 4) {  // 7 exact k-chunks
        v2f a = { buf[(qt * 16 + l16) * 32 + kc + 2 * half],
                  buf[(qt * 16 + l16) * 32 + kc + 2 * half + 1] };
        c = wmma4(a, f_fc2[kc >> 2], c);
      }
      if (l16 < 8) {
        #pragma unroll
        for (int i = 0; i < 8; ++i) {
          int t = qt * 16 + i + 8 * half;
          if (t < SEQT) og[t * 8 + l16] = xb[t * 8 + l16] + c[i];
        }
      }
    }
  }
}

extern "C" void kernel_launch(void* const* d_in, const int* in_sizes, int n_in,
                              void* d_out, int out_size, void* d_ws, size_t ws_size,
                              hipStream_t stream) {
  (void)in_sizes; (void)n_in; (void)out_size; (void)d_ws; (void)ws_size;
  const float* xp     = (const float*)d_in[0];
  const float* ln1_w  = (const float*)d_in[1];
  const float* ln1_b  = (const float*)d_in[2];
  const float* qkv_w  = (const float*)d_in[3];
  const float* proj_w = (const float*)d_in[4];
  const float* ln2_w  = (const float*)d_in[5];
  const float* ln2_b  = (const float*)d_in[6];
  const float* fc1_A  = (const float*)d_in[7];
  const float* fc1_B  = (const float*)d_in[8];
  const float* fc1_Wf = (const float*)d_in[9];
  const float* fc2_A  = (const float*)d_in[10];
  const float* fc2_B  = (const float*)d_in[11];
  const float* fc2_Wf = (const float*)d_in[12];
  float* outp = (float*)d_out;

  dim3 grid(BLK_X, NMODEL);
  dim3 block(NW * 32);
  fused_block_kernel<<<grid, block, 0, stream>>>(
      xp, ln1_w, ln1_b, qkv_w, proj_w, ln2_w, ln2_b,
      fc1_A, fc1_B, fc1_Wf, fc2_A, fc2_B, fc2_Wf, outp);
}